// TemporalContinuityLoss_61813169324393
// MI455X (gfx1250) — compile-verified
//
#include <hip/hip_runtime.h>

// ---------------- problem constants (from reference) ----------------
constexpr int B = 1024;
constexpr int S = 4096;
constexpr int C = 5;

constexpr int THREADS          = 256;                 // 8 waves (wave32)
constexpr int CHUNK_PAIRS      = 2048;                // pairs per block
constexpr int CHUNKS_PER_ROW   = 2;                   // ceil((S-1)/CHUNK_PAIRS)
constexpr int PAIRS_PER_THREAD = CHUNK_PAIRS / THREADS; // 8
constexpr int TILE_T           = CHUNK_PAIRS + 1;     // 2049 timesteps
constexpr int TILE_F           = TILE_T * C;          // 10245 floats
constexpr int TILE_V4          = (TILE_F + 3) / 4;    // 2562 float4 (40992 B)
constexpr int NBLOCKS          = B * CHUNKS_PER_ROW;  // 2048 (multiple of 64)

// CDNA5 async global->LDS path (gfx1250): guarded so the file always compiles.
#if defined(__HIP_DEVICE_COMPILE__) && defined(__gfx1250__) && \
    __has_builtin(__builtin_amdgcn_global_load_async_to_lds_b128)
#define USE_ASYNC_LDS 1
#else
#define USE_ASYNC_LDS 0
#endif

#if USE_ASYNC_LDS
// Builtin parameter types (from clang diagnostic): GCC-vector int4 pointers,
// AS(1) global source, AS(3) LDS destination.
typedef int v4i __attribute__((vector_size(16)));
typedef __attribute__((address_space(1))) v4i GV4;
typedef __attribute__((address_space(3))) v4i LV4;
#endif

// WMMA final reduction (gfx1250 only, guarded).
#if defined(__HIP_DEVICE_COMPILE__) && defined(__gfx1250__) && \
    __has_builtin(__builtin_amdgcn_wmma_f32_16x16x4_f32)
#define USE_WMMA_REDUCE 1
typedef float v2f __attribute__((ext_vector_type(2)));
typedef float v8f __attribute__((ext_vector_type(8)));
#else
#define USE_WMMA_REDUCE 0
#endif

// ---- raw-hardware math helpers (v_exp_f32 / v_log_f32 are base-2) ----
__device__ __forceinline__ float hw_exp2(float x) {
#if defined(__HIP_DEVICE_COMPILE__)
  return __builtin_amdgcn_exp2f(x);
#else
  return exp2f(x);
#endif
}
__device__ __forceinline__ float hw_log2(float x) {
#if defined(__HIP_DEVICE_COMPILE__)
  return __builtin_amdgcn_logf(x);
#else
  return log2f(x);
#endif
}
__device__ __forceinline__ float hw_rcp(float x) {
#if defined(__HIP_DEVICE_COMPILE__)
  return __builtin_amdgcn_rcpf(x);
#else
  return 1.0f / x;
#endif
}

constexpr float LOG2E = 1.4426950408889634f;
constexpr float LN2   = 0.6931471805599453f;

// 5-class stable softmax in base-2: p[c] and lp2[c] = log2(softmax)[c].
__device__ __forceinline__ void softmax5(const float* __restrict__ tile, int t,
                                         float* __restrict__ p,
                                         float* __restrict__ lp2) {
  float u[C];
#pragma unroll
  for (int c = 0; c < C; ++c) u[c] = tile[t * C + c] * LOG2E;
  float m = fmaxf(fmaxf(fmaxf(u[0], u[1]), fmaxf(u[2], u[3])), u[4]);
  float z = 0.f;
#pragma unroll
  for (int c = 0; c < C; ++c) {
    u[c] -= m;
    p[c] = hw_exp2(u[c]);   // v_exp_f32
    z += p[c];
  }
  const float rz  = hw_rcp(z);    // v_rcp_f32
  const float lz2 = hw_log2(z);   // v_log_f32
#pragma unroll
  for (int c = 0; c < C; ++c) {
    p[c] *= rz;
    lp2[c] = u[c] - lz2;          // log2(p[c])
  }
}

__global__ __launch_bounds__(THREADS) void tcl_main_kernel(
    const float* __restrict__ logits, const float* __restrict__ trans,
    float* __restrict__ partial) {
  __shared__ float4 tile4[TILE_V4];
  __shared__ float red[THREADS];
  float* tile = (float*)tile4;

  const int bid   = blockIdx.x;
  const int tid   = threadIdx.x;
  const int chunk = bid & (CHUNKS_PER_ROW - 1);
  const int t0    = chunk * CHUNK_PAIRS;
  const int pairsInChunk = min(CHUNK_PAIRS, (S - 1) - t0);

  // --------- stage: global -> LDS via CDNA5 async b128 copies ---------
  const long long base_f  = (long long)bid * (CHUNK_PAIRS * C);  // 10240 * bid
  const long long total_f = (long long)B * S * C;
  const int nload4 = (int)min((long long)TILE_V4, (total_f - base_f) >> 2);

  const float4* g4 = (const float4*)(logits + base_f);
#if USE_ASYNC_LDS
  for (int i = tid; i < nload4; i += THREADS) {
    __builtin_amdgcn_global_load_async_to_lds_b128(
        (GV4*)(g4 + i), (LV4*)(tile4 + i), 0, 0);
  }
#if __has_builtin(__builtin_amdgcn_s_wait_asynccnt)
  __builtin_amdgcn_s_wait_asynccnt(0);
#else
  asm volatile("s_wait_asynccnt 0" ::: "memory");
#endif
#else
  for (int i = tid; i < nload4; i += THREADS) tile4[i] = g4[i];
#endif
  __syncthreads();

  // --------- transition matrix: uniform address -> scalar loads ---------
  float T[C * C];
#pragma unroll
  for (int i = 0; i < C * C; ++i) T[i] = trans[i];

  // --------- per-thread run of PAIRS_PER_THREAD adjacent pairs ---------
  float acc_sm  = 0.f;  // L1 smoothness sum
  float acc_kl2 = 0.f;  // KL sum in base-2 (scaled by ln2 at the end)
  const int p0 = tid * PAIRS_PER_THREAD;
  if (p0 < pairsInChunk) {
    float pp[C], lpp[C];
    softmax5(tile, p0, pp, lpp);  // prev probs (lpp unused)
#pragma unroll
    for (int k = 0; k < PAIRS_PER_THREAD; ++k) {
      const int pr = p0 + k;
      if (pr >= pairsInChunk) break;
      float pc[C], lpc2[C];
      softmax5(tile, pr + 1, pc, lpc2);

      // smoothness: sum_c |p_prev - p_curr|  (abs folds into add as modifier)
      float sm = 0.f;
#pragma unroll
      for (int c = 0; c < C; ++c) sm += fabsf(pp[c] - pc[c]);
      acc_sm += sm;

      // expected = p_prev @ T ; KL(expected || p_curr) in base-2
#pragma unroll
      for (int d = 0; d < C; ++d) {
        float e = pp[0] * T[0 * C + d];
#pragma unroll
        for (int c = 1; c < C; ++c) e = fmaf(pp[c], T[c * C + d], e);
        if (e > 0.f) acc_kl2 += e * (hw_log2(e) - lpc2[d]);
      }
#pragma unroll
      for (int c = 0; c < C; ++c) pp[c] = pc[c];
    }
  }

  // --------- deterministic block tree reduction ---------
  red[tid] = fmaf(LN2, acc_kl2, acc_sm);
  __syncthreads();
#pragma unroll
  for (int s = THREADS / 2; s > 0; s >>= 1) {
    if (tid < s) red[tid] += red[tid + s];
    __syncthreads();
  }
  if (tid == 0) partial[bid] = red[0];
}

// Final reduction of NBLOCKS partials using the matrix pipe:
// D += A(16x4) x ones(4x16) sums 64 elements per v_wmma_f32_16x16x4_f32.
// Element->A-slot mapping is an arbitrary bijection (fine for a sum).
__global__ __launch_bounds__(32) void tcl_final_kernel(
    const float* __restrict__ partial, float* __restrict__ out, int n) {
#if USE_WMMA_REDUCE
  const int lane = threadIdx.x;      // one full wave32, EXEC all ones
  v8f cacc = {};                     // 16x16 f32 accumulator = 8 VGPRs
  v2f ones; ones.x = 1.f; ones.y = 1.f;
  const int iters = n >> 6;          // n is a multiple of 64
  for (int it = 0; it < iters; ++it) {
    const float2 d = *(const float2*)(partial + (it << 6) + (lane << 1));
    v2f a; a.x = d.x; a.y = d.y;
    cacc = __builtin_amdgcn_wmma_f32_16x16x4_f32(
        false, a, false, ones, (short)0, cacc, false, false);
  }
  // Column n holds all 16 row-sums: M=0..7 on lane n, M=8..15 on lane n+16.
  float s = 0.f;
#pragma unroll
  for (int r = 0; r < 8; ++r) s += cacc[r];
  __shared__ float red[32];
  red[lane] = s;
  __syncthreads();
  if (lane == 0) {
    const double total = (double)red[0] + (double)red[16];
    const double denom = (double)B * (double)(S - 1);
    out[0] = (float)(0.3 * total / denom);  // LAMBDA_SMOOTH == LAMBDA_TRANS
  }
#else
  if (threadIdx.x == 0) {
    double total = 0.0;
    for (int i = 0; i < n; ++i) total += (double)partial[i];
    const double denom = (double)B * (double)(S - 1);
    out[0] = (float)(0.3 * total / denom);
  }
#endif
}

extern "C" void kernel_launch(void* const* d_in, const int* in_sizes, int n_in,
                              void* d_out, int out_size, void* d_ws,
                              size_t ws_size, hipStream_t stream) {
  const float* logits = (const float*)d_in[0];
  const float* trans  = (const float*)d_in[1];
  float* partial      = (float*)d_ws;   // 2048 floats = 8 KB scratch
  float* out          = (float*)d_out;

  tcl_main_kernel<<<NBLOCKS, THREADS, 0, stream>>>(logits, trans, partial);
  tcl_final_kernel<<<1, 32, 0, stream>>>(partial, out, NBLOCKS);
}